// DownsampleUpFIRDn_77859167141932
// MI455X (gfx1250) — compile-verified
//
#include <hip/hip_runtime.h>

// ---- problem constants (from reference: x (8,128,256,256) f32, k 4x4, down=2, pad (1,1))
#define N_       8
#define C_       128
#define H_       256
#define W_       256
#define OH_      128
#define OW_      128
#define TILE_OH  16
#define XROWS    34        // input rows needed per tile = 2*16 + 2
#define HROWS    36        // rounded up to 9 K-chunks of 4 (rows 34,35 zeroed)
#define HSTRIDE  144       // 128 + 16 pad: rows 2 apart land on disjoint LDS bank sets

typedef __attribute__((ext_vector_type(2))) float v2f;
typedef __attribute__((ext_vector_type(8))) float v8f;
typedef __attribute__((ext_vector_type(4))) float f4;
typedef __attribute__((ext_vector_type(4))) int   v4i;

#if defined(__has_builtin)
#  if __has_builtin(__builtin_amdgcn_global_load_async_to_lds_b128)
#    define USE_ASYNC_LDS 1
#  endif
#endif

__global__ __launch_bounds__(256) void
DownsampleUpFIRDn_kernel(const float* __restrict__ x,
                         const float* __restrict__ kern2d,
                         float* __restrict__ out) {
  __shared__ float xs[XROWS * W_];        // 34816 B: raw input tile
  __shared__ float Hs[HROWS * HSTRIDE];   // 20736 B: horizontally filtered, downsampled

  const int t     = threadIdx.x;
  const int tileY = blockIdx.x;           // 0..7
  const int plane = blockIdx.y;           // 0..N_*C_-1
  const int oy0   = tileY * TILE_OH;
  const int row0  = 2 * oy0 - 1;          // pad0 = 1

  // ---- 1D taps (uniform). Row sums of the normalized 2D separable kernel give the
  // normalized 1D kernel; reference does true convolution => flip.
  float kn0 = kern2d[0]  + kern2d[1]  + kern2d[2]  + kern2d[3];
  float kn1 = kern2d[4]  + kern2d[5]  + kern2d[6]  + kern2d[7];
  float kn2 = kern2d[8]  + kern2d[9]  + kern2d[10] + kern2d[11];
  float kn3 = kern2d[12] + kern2d[13] + kern2d[14] + kern2d[15];
  const float t0 = kn3, t1 = kn2, t2 = kn1, t3 = kn0;   // flipped taps

  // ---- zero-fill out-of-bounds input rows (top/bottom pad) and the 2 H pad rows
  for (int r = 0; r < XROWS; ++r) {
    int gr = row0 + r;
    if (gr < 0 || gr >= H_) xs[r * W_ + t] = 0.0f;
  }
  for (int i = t; i < 2 * HSTRIDE; i += 256)
    Hs[(HROWS - 2) * HSTRIDE + i] = 0.0f;

  // ---- stage the input tile: async global->LDS, 16B per lane, each byte read once
  const float* gplane = x + (size_t)plane * (H_ * W_);
  for (int s = t; s < XROWS * (W_ / 4); s += 256) {
    int r  = s >> 6;          // row in tile (W_/4 = 64 b128 segments per row)
    int c4 = s & 63;          // 16-byte column
    int gr = row0 + r;
    if (gr >= 0 && gr < H_) {
      const float* gp = gplane + gr * W_ + c4 * 4;
      float*       lp = xs + r * W_ + c4 * 4;
#if defined(USE_ASYNC_LDS)
      __builtin_amdgcn_global_load_async_to_lds_b128(
          (__attribute__((address_space(1))) v4i*)gp,
          (__attribute__((address_space(3))) v4i*)lp, 0, 0);
#else
      *(f4*)lp = *(const f4*)gp;
#endif
    }
  }
#if defined(USE_ASYNC_LDS)
#  if __has_builtin(__builtin_amdgcn_s_wait_asynccnt)
  __builtin_amdgcn_s_wait_asynccnt(0);
#  else
  asm volatile("s_wait_asynccnt 0" ::: "memory");
#  endif
#endif
  __syncthreads();

  // ---- horizontal FIR + downsample x2 (stride-2 LDS reads are conflict-free in wave32)
  for (int e = t; e < XROWS * OW_; e += 256) {
    int r = e >> 7;
    int c = e & 127;
    const float* xr = xs + r * W_;
    int c0 = 2 * c - 1;
    float v0 = (c0 >= 0)     ? xr[c0]     : 0.0f;   // left pad, col 0 only
    float v1 = xr[c0 + 1];
    float v2 = xr[c0 + 2];
    float v3 = (c0 + 3 < W_) ? xr[c0 + 3] : 0.0f;   // right pad, col 127 only
    Hs[r * HSTRIDE + c] = t0 * v0 + t1 * v1 + t2 * v2 + t3 * v3;
  }
  __syncthreads();

  // ---- vertical FIR + downsample x2 as banded matmul on the fp32 matrix pipe:
  // Out(16x16) = sum_{c=0..8} A_c(16x4) x B_c(4x16),  A_c[m][j] = tap[4c+j-2m]
  const int  wv = t >> 5;         // wave id -> 16-column group
  const int  l  = t & 31;
  const int  m  = l & 15;
  const bool hi = l >= 16;
  const int  n0 = wv * 16;
  const int  j0 = hi ? 2 : 0;     // f32 A/B layout: lanes 16..31 carry K=2,3

  v8f acc = {};
#pragma unroll
  for (int c = 0; c < 9; ++c) {
    int i0 = 4 * c + j0 - 2 * m;
    int i1 = i0 + 1;
    float a0 = 0.0f, a1 = 0.0f;
    a0 = (i0 == 0) ? t0 : a0;  a0 = (i0 == 1) ? t1 : a0;
    a0 = (i0 == 2) ? t2 : a0;  a0 = (i0 == 3) ? t3 : a0;
    a1 = (i1 == 0) ? t0 : a1;  a1 = (i1 == 1) ? t1 : a1;
    a1 = (i1 == 2) ? t2 : a1;  a1 = (i1 == 3) ? t3 : a1;
    v2f a, b;
    a.x = a0;
    a.y = a1;
    b.x = Hs[(4 * c + j0)     * HSTRIDE + n0 + m];
    b.y = Hs[(4 * c + j0 + 1) * HSTRIDE + n0 + m];
    acc = __builtin_amdgcn_wmma_f32_16x16x4_f32(
        /*neg_a=*/false, a, /*neg_b=*/false, b,
        /*c_mod=*/(short)0, acc, /*reuse_a=*/false, /*reuse_b=*/false);
  }

  // ---- store D: lane (m,hi), VGPR v -> out row oy0 + v + (hi?8:0), col n0+m
  float* op = out + (size_t)plane * (OH_ * OW_) + (size_t)oy0 * OW_ + n0 + m;
  const int mbase = hi ? 8 : 0;
#pragma unroll
  for (int v = 0; v < 8; ++v)
    op[(size_t)(v + mbase) * OW_] = acc[v];
}

extern "C" void kernel_launch(void* const* d_in, const int* in_sizes, int n_in,
                              void* d_out, int out_size, void* d_ws, size_t ws_size,
                              hipStream_t stream) {
  const float* x = (const float*)d_in[0];   // (8,128,256,256) f32
  const float* k = (const float*)d_in[1];   // (4,4) f32, sums to 1
  float* out = (float*)d_out;               // (8,128,128,128) f32
  (void)in_sizes; (void)n_in; (void)out_size; (void)d_ws; (void)ws_size;

  dim3 grid(OH_ / TILE_OH, N_ * C_);        // 8 x 1024 workgroups
  DownsampleUpFIRDn_kernel<<<grid, 256, 0, stream>>>(x, k, out);
}